// Model_56805237457211
// MI455X (gfx1250) — compile-verified
//
#include <hip/hip_runtime.h>
#include <hip/hip_bf16.h>
#include <math.h>

typedef __attribute__((ext_vector_type(2))) float v2f;
typedef __attribute__((ext_vector_type(8))) float v8f;

#define TOK_L    4096
#define EMB      50
#define NUM_FEAT 120
#define PADTOK   2
#define NPOS     4107          // 5 + 4096 + 6 padded positions
#define MPAD     4112          // NPOS rounded up to 16
#define NCOLS    1664          // 13*128: 1200 ctx + 400 central + 64 zero pad
#define NREAL    1600
#define NOUT     400
#define MT       257           // MPAD/16
#define MTG      65            // ceil(MT/4)
#define NTG      13            // NCOLS/128
#define KP       26            // 52/2 K-pairs

// ---------------------------------------------------------------- W1 transpose (c-major)
__global__ void k_transpose_w1(const float* __restrict__ W1, float* __restrict__ W1T) {
  int idx = blockIdx.x * blockDim.x + threadIdx.x;       // c*400 + o
  if (idx >= 350 * 400) return;
  int c = idx / 400, o = idx - c * 400;
  W1T[idx] = W1[o * 350 + c];
}

// ---------------------------------------------------------------- zero S
__global__ void k_zero(float* __restrict__ S, int n) {
  int i = blockIdx.x * blockDim.x + threadIdx.x;
  if (i < n) S[i] = 0.0f;
}

// ---------------------------------------------------------------- Apad[p][64]: gathered embeddings, zero K-pad
__global__ void k_build_apad(const int* __restrict__ tokens,
                             const float* __restrict__ words_emb,
                             float* __restrict__ Apad) {
  int idx = blockIdx.x * blockDim.x + threadIdx.x;
  if (idx >= MPAD * 64) return;
  int p = idx >> 6, d = idx & 63;
  int tok = (p >= 5 && p < 5 + TOK_L) ? tokens[p - 5] : PADTOK;
  Apad[idx] = (d < EMB) ? words_emb[tok * EMB + d] : 0.0f;
}

// ---------------------------------------------------------------- BmatP[(kpair*NCOLS + n)*2 + e] = B[2*kpair+e][n]
// B[k][n]: n<1200 -> W1T[(50*(n/400)+k)*400 + n%400]; 1200<=n<1600 -> W1T[(300+k)*400 + n-1200]; else 0
__global__ void k_build_bmatp(const float* __restrict__ W1T, float* __restrict__ BmatP) {
  int idx = blockIdx.x * blockDim.x + threadIdx.x;
  if (idx >= KP * NCOLS * 2) return;
  int e = idx & 1;
  int n = (idx >> 1) % NCOLS;
  int kpair = (idx >> 1) / NCOLS;
  int k = 2 * kpair + e;
  float v = 0.0f;
  if (k < EMB && n < NREAL) {
    int c, o;
    if (n < 1200) { int j = n / 400; o = n - j * 400; c = 50 * j + k; }
    else          { o = n - 1200;    c = 300 + k; }
    v = W1T[c * 400 + o];
  }
  BmatP[idx] = v;
}

// ---------------------------------------------------------------- WMMA GEMM: pdot = Apad x B  (M=4112, N=1664, K=52)
// 128 threads = 4 waves; block stages a 52x128 B strip into LDS (K-pair layout),
// each wave computes a 16x128 output strip: 104 unconditional WMMAs.
__global__ void k_pdot_gemm(const float* __restrict__ Apad,
                            const float* __restrict__ BmatP,
                            float* __restrict__ pdot) {
  __shared__ __align__(16) float Bsl[KP * 256];          // 26 kpairs x 128 cols x 2
  int t = threadIdx.x;
  int job = blockIdx.x;
  int mtg = job % MTG;
  int ntg = job / MTG;

  // stage B strip: contiguous float4 copy, no guards
  {
    const float4* src = reinterpret_cast<const float4*>(BmatP) + ntg * 64;
    float4* dst = reinterpret_cast<float4*>(Bsl);
    for (int i4 = t; i4 < KP * 64; i4 += 128) {
      int kpair = i4 >> 6, r4 = i4 & 63;
      dst[kpair * 64 + r4] = src[kpair * 832 + r4];      // 832 = NCOLS*2/4
    }
  }
  __syncthreads();

  int wv = t >> 5, lane = t & 31;
  int mt = mtg * 4 + wv;
  if (mt < MT) {                                          // wave-uniform, after barrier
    int g = lane >> 4, ln = lane & 15;

    // preload all A fragments for this wave's 16 rows (one clause of b64 loads)
    const float* arow = Apad + (size_t)(mt * 16 + ln) * 64 + 2 * g;
    v2f areg[13];
#pragma unroll
    for (int kt = 0; kt < 13; ++kt)
      areg[kt] = *reinterpret_cast<const v2f*>(arow + 4 * kt);

    v8f acc[8];
    v8f zero8 = {0.f, 0.f, 0.f, 0.f, 0.f, 0.f, 0.f, 0.f};
#pragma unroll
    for (int i = 0; i < 8; ++i) acc[i] = zero8;

#pragma unroll
    for (int kt = 0; kt < 13; ++kt) {
      const v2f* brow = reinterpret_cast<const v2f*>(Bsl) + (2 * kt + g) * 128 + ln;
#pragma unroll
      for (int nt8 = 0; nt8 < 8; ++nt8) {
        v2f b = brow[nt8 * 16];                           // ds_load_b64, conflict-free
        acc[nt8] = __builtin_amdgcn_wmma_f32_16x16x4_f32(
            false, areg[kt], false, b, (short)0, acc[nt8], false, false);
      }
    }

    // store D: element r of lane (ln,g) is C[M = r + 8*g][N = ln]
    int nbase = ntg * 128 + ln;
#pragma unroll
    for (int nt8 = 0; nt8 < 8; ++nt8) {
#pragma unroll
      for (int r = 0; r < 8; ++r)
        pdot[(size_t)(mt * 16 + r + 8 * g) * NCOLS + nbase + nt8 * 16] = acc[nt8][r];
    }
  }
}

// ---------------------------------------------------------------- posdot[f][o]
__global__ void k_posdot(const float* __restrict__ pos_emb,
                         const float* __restrict__ W1T,
                         float* __restrict__ posdot) {
  int f = blockIdx.x;
  int rx = 0, ry = 1, rz = 2, cnt = 0;
  for (int x = 0; x < 10 && cnt <= f; ++x)
    for (int y = x + 1; y < 10 && cnt <= f; ++y)
      for (int z = y + 1; z < 10 && cnt <= f; ++z) {
        if (cnt == f) { rx = x; ry = y; rz = z; }
        ++cnt;
      }
  const float* ex = pos_emb + rx * EMB;
  const float* ey = pos_emb + ry * EMB;
  const float* ez = pos_emb + rz * EMB;
  for (int o = threadIdx.x; o < NOUT; o += blockDim.x) {
    float acc = 0.0f;
    for (int d = 0; d < EMB; ++d) {
      acc = fmaf(ex[d], W1T[(150 + d) * 400 + o], acc);
      acc = fmaf(ey[d], W1T[(200 + d) * 400 + o], acc);
      acc = fmaf(ez[d], W1T[(250 + d) * 400 + o], acc);
    }
    posdot[f * NOUT + o] = acc;
  }
}

// ---------------------------------------------------------------- combine: S[o] += sum_f wfeat[f]*relu(z[l,f,o])
// 416 threads (13 waves), one o per thread; 10 pdot rows cached in LDS.
__global__ void k_combine(const float* __restrict__ pdot,
                          const float* __restrict__ posdot,
                          const float* __restrict__ b1,
                          const float* __restrict__ wfeat,
                          float* __restrict__ S) {
  __shared__ __align__(16) float lctx[10 * 1200];  // rows p=l+1..l+10, cols 0..1199
  __shared__ float cdb[NOUT];
  __shared__ float wf[NUM_FEAT];
  __shared__ int   rx[NUM_FEAT], ry[NUM_FEAT], rz[NUM_FEAT];

  int l = blockIdx.x;
  int t = threadIdx.x;

  for (int i = t; i < 3000; i += 416) {            // 10 rows x 300 float4
    int pi = i / 300, rem = i - pi * 300;
    const float4 s = reinterpret_cast<const float4*>(pdot + (size_t)(l + 1 + pi) * NCOLS)[rem];
    reinterpret_cast<float4*>(lctx + pi * 1200)[rem] = s;
  }
  for (int o = t; o < NOUT; o += 416)
    cdb[o] = pdot[(size_t)(l + 5) * NCOLS + 1200 + o] + b1[o];
  if (t < NUM_FEAT) wf[t] = wfeat[t];
  if (t == 416 - 1) {
    int f = 0;
    for (int x = 0; x < 10; ++x)
      for (int y = x + 1; y < 10; ++y)
        for (int z = y + 1; z < 10; ++z) { rx[f] = x; ry[f] = y; rz[f] = z; ++f; }
  }
  __syncthreads();

  if (t < NOUT) {
    int o = t;
    float acc = 0.0f;
    float c0 = cdb[o];
    for (int f = 0; f < NUM_FEAT; ++f) {
      float s0 = lctx[rx[f] * 1200 + o]
               + lctx[ry[f] * 1200 + 400 + o]
               + lctx[rz[f] * 1200 + 800 + o]
               + posdot[f * NOUT + o] + c0;
      acc = fmaf(wf[f], fmaxf(s0, 0.0f), acc);
    }
    unsafeAtomicAdd(&S[o], acc);                   // global_atomic_add_f32
  }
}

// ---------------------------------------------------------------- finalize: loss
__global__ void k_finalize(const float* __restrict__ S,
                           const float* __restrict__ W2,
                           const float* __restrict__ b2,
                           const float* __restrict__ wfeat,
                           const int* __restrict__ target,
                           float* __restrict__ out) {
  int lane = threadIdx.x;                          // one wave32
  float d0 = 0.0f, d1 = 0.0f, swf = 0.0f;
  for (int o = lane; o < NOUT; o += 32) {
    float s = S[o];
    d0 = fmaf(W2[o], s, d0);
    d1 = fmaf(W2[400 + o], s, d1);
  }
  for (int f = lane; f < NUM_FEAT; f += 32) swf += wfeat[f];
  for (int off = 16; off > 0; off >>= 1) {
    d0  += __shfl_down(d0, off, 32);
    d1  += __shfl_down(d1, off, 32);
    swf += __shfl_down(swf, off, 32);
  }
  if (lane == 0) {
    const float inv = 1.0f / ((float)TOK_L * (float)NUM_FEAT);
    float w0 = (d0 + (float)TOK_L * swf * b2[0]) * inv;
    float w1 = (d1 + (float)TOK_L * swf * b2[1]) * inv;
    float m = fmaxf(w0, w1);
    float lse = m + logf(expf(w0 - m) + expf(w1 - m));
    float wt = (target[0] == 0) ? w0 : w1;
    out[0] = lse - wt;
  }
}

// ---------------------------------------------------------------- launcher
extern "C" void kernel_launch(void* const* d_in, const int* in_sizes, int n_in,
                              void* d_out, int out_size, void* d_ws, size_t ws_size,
                              hipStream_t stream) {
  const int*   tokens    = (const int*)d_in[0];
  const int*   target    = (const int*)d_in[1];
  const float* words_emb = (const float*)d_in[2];
  const float* pos_emb   = (const float*)d_in[3];
  const float* W1        = (const float*)d_in[4];
  const float* b1        = (const float*)d_in[5];
  const float* W2        = (const float*)d_in[6];
  const float* b2        = (const float*)d_in[7];
  const float* wfeat     = (const float*)d_in[8];
  float* out = (float*)d_out;

  float* ws     = (float*)d_ws;
  float* W1T    = ws;                                   // 350*400        = 140000
  float* Apad   = W1T + 350 * 400;                      // MPAD*64        = 263168
  float* BmatP  = Apad + (size_t)MPAD * 64;             // KP*NCOLS*2     = 86528
  float* pdot   = BmatP + (size_t)KP * NCOLS * 2;       // MPAD*NCOLS     = 6842368
  float* posdot = pdot + (size_t)MPAD * NCOLS;          // 120*400        = 48000
  float* S      = posdot + NUM_FEAT * NOUT;             // 400

  k_transpose_w1<<<(350 * 400 + 255) / 256, 256, 0, stream>>>(W1, W1T);
  k_build_apad<<<(MPAD * 64 + 255) / 256, 256, 0, stream>>>(tokens, words_emb, Apad);
  k_build_bmatp<<<(KP * NCOLS * 2 + 255) / 256, 256, 0, stream>>>(W1T, BmatP);
  k_zero<<<2, 256, 0, stream>>>(S, NOUT);

  k_pdot_gemm<<<MTG * NTG, 128, 0, stream>>>(Apad, BmatP, pdot);

  k_posdot<<<NUM_FEAT, 256, 0, stream>>>(pos_emb, W1T, posdot);
  k_combine<<<TOK_L, 416, 0, stream>>>(pdot, posdot, b1, wfeat, S);
  k_finalize<<<1, 32, 0, stream>>>(S, W2, b2, wfeat, target, out);
}